// GAT_79894981640316
// MI455X (gfx1250) — compile-verified
//
#include <hip/hip_runtime.h>
#include <math.h>

#define N_NODES 100000
#define N_EDGES 1600000
#define IN_DIM 128
#define HEADS 4
#define HID 32
#define HC 128
#define NUM_CLASSES 349
#define BN_EPS 1e-5f
#define NEG_SLOPE 0.2f

typedef __attribute__((ext_vector_type(2))) float v2f;
typedef __attribute__((ext_vector_type(8))) float v8f;

// ---------------------------------------------------------------------------
// WMMA GEMM: out[N_NODES, NCOLS] = x[N_NODES,128] @ W[128,NCOLS] (+ bias)
// block = 128 threads (4 waves). Block owns 16 rows of x staged in LDS
// (row stride padded to 130 floats -> conflict-free ds_load_b64); each wave
// computes one 16x16 output tile with 32x v_wmma_f32_16x16x4_f32 (exact f32).
// NCOLS is a template constant so all W/out addressing folds into immediate
// offsets; the ragged-N guard is a branch-free cndmask on the B fragment.
// A frag layout (ISA 7.12.2): lanes 0-15 hold K=k,k+1; lanes 16-31 K=k+2,k+3.
// B frag mirror: lane&15 carries column N, (lane>>4) selects the K pair.
// ---------------------------------------------------------------------------
template <int NCOLS>
__global__ __launch_bounds__(128) void gemm_x_w(
    const float* __restrict__ x, const float* __restrict__ W,
    const float* __restrict__ bias, float* __restrict__ out) {
  __shared__ float lx[16 * 130];
  const int lane = threadIdx.x & 31;
  const int wave = threadIdx.x >> 5;
  const int m0 = blockIdx.y * 16;
  const int n0 = blockIdx.x * 64 + wave * 16;

  for (int i = threadIdx.x; i < 16 * 128; i += 128) {
    int r = i >> 7, c = i & 127;
    lx[r * 130 + c] = x[(size_t)(m0 + r) * IN_DIM + c];
  }
  __syncthreads();

  const int arow = lane & 15;
  const int koff = (lane >> 4) * 2;          // 0 for lanes 0-15, 2 for 16-31
  const int bcol = n0 + (lane & 15);
  const bool colok = (NCOLS % 64 == 0) ? true : (bcol < NCOLS);
  const int bc = colok ? bcol : 0;           // clamped: loads stay in-bounds
  const float* __restrict__ Wp = W + bc;

  v8f acc = {};
#pragma unroll
  for (int k = 0; k < IN_DIM; k += 4) {
    v2f a, b;
    a.x = lx[arow * 130 + k + koff];
    a.y = lx[arow * 130 + k + koff + 1];
    float w0 = Wp[(size_t)(k + koff) * NCOLS];       // immediate offsets
    float w1 = Wp[(size_t)(k + koff + 1) * NCOLS];
    b.x = colok ? w0 : 0.0f;                         // v_cndmask, no branch
    b.y = colok ? w1 : 0.0f;
    acc = __builtin_amdgcn_wmma_f32_16x16x4_f32(
        /*neg_a=*/false, a, /*neg_b=*/false, b,
        /*c_mod=*/(short)0, acc, /*reuse_a=*/false, /*reuse_b=*/false);
  }
  if (!colok) return;
  const float bv = bias ? bias[bcol] : 0.0f;
  const int rbase = m0 + ((lane >> 4) * 8);  // D layout: lanes16-31 -> M=r+8
#pragma unroll
  for (int r = 0; r < 8; ++r)
    out[(size_t)(rbase + r) * NCOLS + bcol] = acc[r] + bv;
}

// s_src[n,h] = dot(xs[n,h,:], a_src[h,:]);  s_dst likewise
__global__ void scores_k(const float* __restrict__ xs, const float* __restrict__ xd,
                         const float* __restrict__ a_src, const float* __restrict__ a_dst,
                         float* __restrict__ s_src, float* __restrict__ s_dst) {
  int gid = blockIdx.x * blockDim.x + threadIdx.x;
  if (gid >= N_NODES * HEADS) return;
  int n = gid >> 2, h = gid & 3;
  const float* pxs = xs + (size_t)n * HC + h * HID;
  const float* pxd = xd + (size_t)n * HC + h * HID;
  const float* pas = a_src + h * HID;
  const float* pad = a_dst + h * HID;
  float ss = 0.f, sd = 0.f;
#pragma unroll
  for (int j = 0; j < HID; ++j) { ss += pxs[j] * pas[j]; sd += pxd[j] * pad[j]; }
  s_src[gid] = ss; s_dst[gid] = sd;
}

__global__ void init_k(float* __restrict__ mx, float* __restrict__ denom,
                       float* __restrict__ sums, float* __restrict__ sumsq) {
  int gid = blockIdx.x * blockDim.x + threadIdx.x;
  if (gid < N_NODES * HEADS) { mx[gid] = -INFINITY; denom[gid] = 0.f; }
  if (gid < HC) { sums[gid] = 0.f; sumsq[gid] = 0.f; }
}

// pass 1: e = leaky_relu(s_src[src]+s_dst[dst]); segment max via ordered-int
// float atomic-max trick (positive -> atomicMax(int), negative -> atomicMin(uint))
__global__ void edge_max_k(const int* __restrict__ src, const int* __restrict__ dst,
                           const float* __restrict__ s_src, const float* __restrict__ s_dst,
                           float* __restrict__ ebuf, float* __restrict__ mx) {
  int gid = blockIdx.x * blockDim.x + threadIdx.x;
  if (gid >= N_EDGES * HEADS) return;
  int e = gid >> 2, h = gid & 3;
  int s = src[e], d = dst[e];
  float v = s_src[s * HEADS + h] + s_dst[d * HEADS + h];
  v = v > 0.f ? v : NEG_SLOPE * v;
  ebuf[gid] = v;
  float* addr = &mx[d * HEADS + h];
  if (v >= 0.f) atomicMax((int*)addr, __float_as_int(v));
  else          atomicMin((unsigned int*)addr, __float_as_uint(v));
}

// pass 2: p = exp(e - m[dst]); denom[dst] += p; store p in place
__global__ void edge_den_k(const int* __restrict__ dst, const float* __restrict__ mx,
                           float* __restrict__ ebuf, float* __restrict__ denom) {
  int gid = blockIdx.x * blockDim.x + threadIdx.x;
  if (gid >= N_EDGES * HEADS) return;
  int e = gid >> 2, h = gid & 3;
  int d = dst[e];
  float m = mx[d * HEADS + h];
  if (m < -3.0e38f) m = 0.f;              // isolated-node guard (isfinite)
  float p = expf(ebuf[gid] - m);
  ebuf[gid] = p;
  atomicAdd(&denom[d * HEADS + h], p);
}

// pass 3: one wave per edge, 4 channels per lane: agg[dst] += xs[src]*alpha
__global__ __launch_bounds__(256) void edge_agg_k(
    const int* __restrict__ src, const int* __restrict__ dst,
    const float* __restrict__ ebuf, const float* __restrict__ denom,
    const float* __restrict__ xs, float* __restrict__ agg) {
  const int lane = threadIdx.x & 31;
  const int wave = threadIdx.x >> 5;
  const long e = (long)blockIdx.x * 8 + wave;
  if (e >= N_EDGES) return;
  const int s = src[e], d = dst[e];
  __builtin_prefetch(xs + (size_t)s * HC, 0, 3);      // global_prefetch_b8
  const int h = lane >> 3;                             // 8 lanes per head
  const float p = ebuf[e * HEADS + h];
  const float den = denom[d * HEADS + h];
  const float alpha = p / (den + 1e-16f);
  const float4 v = *reinterpret_cast<const float4*>(xs + (size_t)s * HC + lane * 4);
  float* o = agg + (size_t)d * HC + lane * 4;
  atomicAdd(o + 0, v.x * alpha);
  atomicAdd(o + 1, v.y * alpha);
  atomicAdd(o + 2, v.z * alpha);
  atomicAdd(o + 3, v.w * alpha);
}

// h = agg + conv_bias + skip (in place); per-channel sum / sumsq reduction
__global__ __launch_bounds__(128) void bn_stats_k(
    float* __restrict__ agg, const float* __restrict__ skip,
    const float* __restrict__ conv_bias, float* __restrict__ sums,
    float* __restrict__ sumsq) {
  const int c = threadIdx.x;
  const int row0 = blockIdx.x * 256;
  const float cb = conv_bias[c];
  float sum = 0.f, sq = 0.f;
  for (int r = 0; r < 256; ++r) {
    int row = row0 + r;
    if (row < N_NODES) {
      size_t idx = (size_t)row * HC + c;
      float hv = agg[idx] + cb + skip[idx];
      agg[idx] = hv;
      sum += hv; sq += hv * hv;
    }
  }
  atomicAdd(&sums[c], sum);
  atomicAdd(&sumsq[c], sq);
}

// x_out = relu(gamma*(h-mu)*rsqrt(var+eps)+beta), biased variance
__global__ void bn_apply_k(const float* __restrict__ agg, const float* __restrict__ sums,
                           const float* __restrict__ sumsq, const float* __restrict__ gamma,
                           const float* __restrict__ beta, float* __restrict__ xout) {
  int gid = blockIdx.x * blockDim.x + threadIdx.x;
  if (gid >= N_NODES * HC) return;
  int c = gid & 127;
  const float inv_n = 1.0f / (float)N_NODES;
  float mu = sums[c] * inv_n;
  float var = sumsq[c] * inv_n - mu * mu;
  float y = gamma[c] * (agg[gid] - mu) * rsqrtf(var + BN_EPS) + beta[c];
  xout[gid] = y > 0.f ? y : 0.f;
}

extern "C" void kernel_launch(void* const* d_in, const int* in_sizes, int n_in,
                              void* d_out, int out_size, void* d_ws, size_t ws_size,
                              hipStream_t stream) {
  const float* x         = (const float*)d_in[0];
  const int*   edge_idx  = (const int*)d_in[1];
  const float* W_src     = (const float*)d_in[2];
  const float* W_dst     = (const float*)d_in[3];
  const float* att_src   = (const float*)d_in[4];
  const float* att_dst   = (const float*)d_in[5];
  const float* conv_bias = (const float*)d_in[6];
  const float* lin_W     = (const float*)d_in[7];
  const float* lin_b     = (const float*)d_in[8];
  const float* bn_gamma  = (const float*)d_in[9];
  const float* bn_beta   = (const float*)d_in[10];
  const float* fc_W      = (const float*)d_in[11];
  const float* fc_b      = (const float*)d_in[12];

  const int* src = edge_idx;
  const int* dst = edge_idx + N_EDGES;

  const size_t NBf = (size_t)N_NODES * HC;   // 12.8M floats
  float* ws    = (float*)d_ws;
  float* xs    = ws;
  float* xd    = xs + NBf;
  float* skip  = xd + NBf;
  float* agg   = skip + NBf;
  float* xbuf  = agg + NBf;
  float* s_src = xbuf + NBf;
  float* s_dst = s_src + (size_t)N_NODES * HEADS;
  float* mx    = s_dst + (size_t)N_NODES * HEADS;
  float* denom = mx + (size_t)N_NODES * HEADS;
  float* ebuf  = denom + (size_t)N_NODES * HEADS;
  float* sums  = ebuf + (size_t)N_EDGES * HEADS;
  float* sumsq = sums + HC;

  const dim3 gemm_grid(2, N_NODES / 16);          // 128 cols, 16-row tiles
  const int nh_blocks = (N_NODES * HEADS + 255) / 256;
  const int eh_blocks = (N_EDGES * HEADS + 255) / 256;

  for (int layer = 0; layer < 2; ++layer) {
    const float* xin = (layer == 0) ? x : xbuf;
    const size_t wo = (size_t)layer * IN_DIM * HC;

    gemm_x_w<HC><<<gemm_grid, 128, 0, stream>>>(xin, W_src + wo, nullptr, xs);
    gemm_x_w<HC><<<gemm_grid, 128, 0, stream>>>(xin, W_dst + wo, nullptr, xd);
    gemm_x_w<HC><<<gemm_grid, 128, 0, stream>>>(xin, lin_W + wo, lin_b + layer * HC, skip);

    scores_k<<<nh_blocks, 256, 0, stream>>>(xs, xd, att_src + layer * HC,
                                            att_dst + layer * HC, s_src, s_dst);
    init_k<<<nh_blocks, 256, 0, stream>>>(mx, denom, sums, sumsq);
    hipMemsetAsync(agg, 0, NBf * sizeof(float), stream);

    edge_max_k<<<eh_blocks, 256, 0, stream>>>(src, dst, s_src, s_dst, ebuf, mx);
    edge_den_k<<<eh_blocks, 256, 0, stream>>>(dst, mx, ebuf, denom);
    edge_agg_k<<<(N_EDGES + 7) / 8, 256, 0, stream>>>(src, dst, ebuf, denom, xs, agg);

    bn_stats_k<<<(N_NODES + 255) / 256, 128, 0, stream>>>(agg, skip,
                                                          conv_bias + layer * HC, sums, sumsq);
    bn_apply_k<<<(N_NODES * HC + 255) / 256, 256, 0, stream>>>(
        agg, sums, sumsq, bn_gamma + layer * HC, bn_beta + layer * HC, xbuf);
  }

  // final classifier: [N,128] @ [128,349] + b  (ragged N guarded in-kernel)
  gemm_x_w<NUM_CLASSES><<<dim3((NUM_CLASSES + 63) / 64, N_NODES / 16), 128, 0, stream>>>(
      xbuf, fc_W, fc_b, (float*)d_out);
}